// CombinedGATConv_43430709297253
// MI455X (gfx1250) — compile-verified
//
#include <hip/hip_runtime.h>
#include <hip/hip_bf16.h>
#include <math.h>

// ---------------------------------------------------------------------------
// GAT layer (CombinedGATConv) for MI455X / gfx1250, wave32.
//   content GEMM   -> v_wmma_f32_16x16x4_f32, operands staged in LDS via
//                     TDM (tensor_load_to_lds, weights, LDS row padding) and
//                     global_load_async_to_lds_b128 (per-wave A tile).
//   edge softmax   -> int-keyed atomicMax + atomicAdd (L2-resident)
//   output scatter -> global_atomic_add_f32 (content fits in 192MB L2)
// ---------------------------------------------------------------------------

typedef float v2f __attribute__((ext_vector_type(2)));
typedef float v8f __attribute__((ext_vector_type(8)));
typedef unsigned int u32x4 __attribute__((ext_vector_type(4)));
typedef unsigned int u32x8 __attribute__((ext_vector_type(8)));

#define N_NODES   100000
#define IN_CH     128
#define POS       16
#define ROWLEN    (IN_CH + POS)   // 144
#define H_HEADS   4
#define C_CH      16
#define HC        (H_HEADS * C_CH) // 64
#define NEG_SLOPE 0.2f
#define DIV_W     0.1f
// LDS row stride: 128 data floats + 4 pad floats -> 528B rows, bank advance 4
// per row => half-wave lanes m=0..15 hit banks {4m,4m+1}: conflict-free b64.
#define LROW      (IN_CH + 4)      // 132

// order-preserving float<->int key for atomicMax on signed int
__device__ __forceinline__ int f2key(float f) {
    int b = __float_as_int(f);
    return b >= 0 ? b : (b ^ 0x7fffffff);
}
__device__ __forceinline__ float key2f(int k) {
    int b = k >= 0 ? k : (k ^ 0x7fffffff);
    return __int_as_float(b);
}

// ---------------------------------------------------------------------------
// Kernel 0: initialize accumulators. out = bias broadcast, amax = -inf key,
// asum = 0, gram = 0.
// ---------------------------------------------------------------------------
__global__ void init_kernel(const float* __restrict__ bias,
                            float* __restrict__ out,
                            int* __restrict__ amaxkey,
                            float* __restrict__ asum,
                            float* __restrict__ gram,
                            int n_nodes) {
    int i = blockIdx.x * blockDim.x + threadIdx.x;
    int total = n_nodes * HC;
    if (i < total)        out[i] = bias[i & (HC - 1)];
    if (i < n_nodes * H_HEADS) { amaxkey[i] = (int)0x80000000; asum[i] = 0.0f; }
    if (i < 16)           gram[i] = 0.0f;
}

// ---------------------------------------------------------------------------
// Kernel 1: content = x[:, :128] @ lin_w.T  via V_WMMA_F32_16X16X4_F32.
//
// Data movement (CDNA5 paths):
//  * Weights (64x128 f32, row-major): one TDM descriptor, wave 0 issues
//    tensor_load_to_lds with pad_enable (4 dwords pad per 128 dwords) so the
//    LDS image lands with a 132-float row stride. Waited with TENSORcnt.
//  * A tile (16 rows x 128 f32 per wave): 16x global_load_async_to_lds_b128,
//    coalesced float4 pattern, padded rows in LDS. Waited with ASYNCcnt.
//
// WMMA fragment maps (wave32):
//  A frag (16x4):  lane L -> m = L&15, kpair = (L>>4)*2, vgpr = k%2
//  B frag (4x16):  lane L -> n = L&15, kpair = (L>>4)*2, vgpr = k%2
//  D tile (16x16): vgpr r -> m = (L>>4)*8 + r, n = L&15
// ---------------------------------------------------------------------------
__global__ __launch_bounds__(128) void gemm_content_kernel(
        const float* __restrict__ x,
        const float* __restrict__ lin_w,
        float* __restrict__ content,
        int n_nodes) {
    __shared__ float wlds[HC * LROW];      // 64 x 132 floats = 33 KB
    __shared__ float alds[4][16 * LROW];   // 4 waves x 16 x 132 floats = 33 KB

    const int tid  = threadIdx.x;
    const int wave = tid >> 5;
    const int lane = tid & 31;

    // ---- wave 0: TDM load of the whole weight matrix into LDS -------------
    if (wave == 0) {
        unsigned long long ga = (unsigned long long)(uintptr_t)lin_w;
        u32x4 g0;
        g0[0] = 1u;                                   // count=1, user mode
        g0[1] = (unsigned)(uintptr_t)&wlds[0];        // lds_addr (byte offset)
        g0[2] = (unsigned)ga;                         // global_addr[31:0]
        g0[3] = (unsigned)(ga >> 32) | (2u << 30);    // addr[56:32] | type=2
        u32x8 g1;
        g1[0] = (2u << 16)          // data_size = 4B
              | (1u << 20)          // pad_enable
              | (6u << 22)          // pad_interval: 128 dwords
              | (3u << 25);         // pad_amount:   4 dwords
        g1[1] = (IN_CH & 0xffffu) << 16;                      // tensor_dim0 lo16
        g1[2] = (IN_CH >> 16) | ((HC & 0xffffu) << 16);       // dim0 hi | dim1 lo
        g1[3] = (HC >> 16) | ((unsigned)IN_CH << 16);         // dim1 hi | tile_dim0
        g1[4] = HC;                                           // tile_dim1 | tile_dim2=0
        g1[5] = IN_CH;                                        // tensor_dim0_stride lo32
        g1[6] = 0u;                                           // stride hi | dim1_stride lo
        g1[7] = 0u;
        asm volatile("tensor_load_to_lds %0, %1" :: "s"(g0), "s"(g1) : "memory");
        __builtin_amdgcn_s_wait_tensorcnt(0);
    }
    __syncthreads();   // publish weight tile to all waves

    const int strip   = blockIdx.x * 4 + wave;     // 16-row strip id
    const int nstrips = (n_nodes + 15) >> 4;
    if (strip >= nstrips) return;
    const int row0 = strip << 4;

    // ---- per-wave async stage of the 16x128 A tile into padded LDS --------
    {
        const unsigned abase = (unsigned)(uintptr_t)&alds[wave][0];
        #pragma unroll
        for (int it = 0; it < 16; ++it) {
            int idx = it * 32 + lane;        // float4 index within tile, 0..511
            int r   = idx >> 5;              // row 0..15
            int c4  = idx & 31;              // float4 column 0..31
            unsigned lds_off = abase + (unsigned)(r * LROW + c4 * 4) * 4u;
            unsigned long long gaddr =
                (unsigned long long)(uintptr_t)(x + (size_t)(row0 + r) * ROWLEN + c4 * 4);
            asm volatile("global_load_async_to_lds_b128 %0, %1, off"
                         :: "v"(lds_off), "v"(gaddr) : "memory");
        }
        asm volatile("s_wait_asynccnt 0x0" ::: "memory");
    }

    const int m  = lane & 15;
    const int kh = (lane >> 4) << 1;               // 0 or 2

    const float* arow = &alds[wave][m * LROW + kh];
    const float* bb   = &wlds[m * LROW + kh];

    v8f acc0 = {}, acc1 = {}, acc2 = {}, acc3 = {};

    #pragma unroll
    for (int k0 = 0; k0 < IN_CH; k0 += 4) {
        v2f a;  a.x  = arow[k0];                a.y  = arow[k0 + 1];
        v2f b0; b0.x = bb[k0];                  b0.y = bb[k0 + 1];
        v2f b1; b1.x = bb[k0 + 16 * LROW];      b1.y = bb[k0 + 16 * LROW + 1];
        v2f b2; b2.x = bb[k0 + 32 * LROW];      b2.y = bb[k0 + 32 * LROW + 1];
        v2f b3; b3.x = bb[k0 + 48 * LROW];      b3.y = bb[k0 + 48 * LROW + 1];
        acc0 = __builtin_amdgcn_wmma_f32_16x16x4_f32(false, a, false, b0, (short)0, acc0, false, false);
        acc1 = __builtin_amdgcn_wmma_f32_16x16x4_f32(false, a, false, b1, (short)0, acc1, false, false);
        acc2 = __builtin_amdgcn_wmma_f32_16x16x4_f32(false, a, false, b2, (short)0, acc2, false, false);
        acc3 = __builtin_amdgcn_wmma_f32_16x16x4_f32(false, a, false, b3, (short)0, acc3, false, false);
    }

    const int mbase = (lane >> 4) << 3;            // 0 or 8
    const int n     = lane & 15;
    float* dst = content + (size_t)(row0 + mbase) * HC + n;
    #pragma unroll
    for (int r = 0; r < 8; ++r) {
        dst[(size_t)r * HC +  0] = acc0[r];
        dst[(size_t)r * HC + 16] = acc1[r];
        dst[(size_t)r * HC + 32] = acc2[r];
        dst[(size_t)r * HC + 48] = acc3[r];
    }
}

// ---------------------------------------------------------------------------
// Kernel 2: per-(node, head) attention scalars:
//   s[n,h] = content[n,h,:]·att[h,:C]  + pos[n,:]·pos_att[h,:POS]    (row side)
//   t[n,h] = content[n,h,:]·att[h,C:]  + pos[n,:]·pos_att[h,POS:]    (col side)
// ---------------------------------------------------------------------------
__global__ void node_st_kernel(const float* __restrict__ x,
                               const float* __restrict__ content,
                               const float* __restrict__ att,
                               const float* __restrict__ pos_att,
                               float* __restrict__ svec,
                               float* __restrict__ tvec,
                               int n_nodes) {
    int i = blockIdx.x * blockDim.x + threadIdx.x;
    if (i >= n_nodes * H_HEADS) return;
    int nd = i >> 2, h = i & 3;

    const float* c  = content + (size_t)nd * HC + h * C_CH;
    const float* al = att + h * 2 * C_CH;
    float s = 0.f, t = 0.f;
    #pragma unroll
    for (int k = 0; k < C_CH; ++k) {
        float cv = c[k];
        s = fmaf(cv, al[k], s);
        t = fmaf(cv, al[C_CH + k], t);
    }
    const float* p  = x + (size_t)nd * ROWLEN + IN_CH;
    const float* pl = pos_att + h * 2 * POS;
    #pragma unroll
    for (int k = 0; k < POS; ++k) {
        float pv = p[k];
        s = fmaf(pv, pl[k], s);
        t = fmaf(pv, pl[POS + k], t);
    }
    svec[i] = s;
    tvec[i] = t;
}

// ---------------------------------------------------------------------------
// Kernel 3: edge pass 1 — alpha = leaky_relu(s[row]+t[col]); segment max.
// ---------------------------------------------------------------------------
__global__ void edge_alpha_kernel(const int* __restrict__ eidx, int E, int Etot,
                                  const float* __restrict__ svec,
                                  const float* __restrict__ tvec,
                                  float* __restrict__ alpha,
                                  int* __restrict__ amaxkey) {
    int e = blockIdx.x * blockDim.x + threadIdx.x;
    if (e >= Etot) return;
    int r = (e < E) ? eidx[e]     : (e - E);
    int c = (e < E) ? eidx[E + e] : (e - E);
    #pragma unroll
    for (int h = 0; h < H_HEADS; ++h) {
        float a = svec[r * H_HEADS + h] + tvec[c * H_HEADS + h];
        a = a > 0.0f ? a : NEG_SLOPE * a;
        alpha[(size_t)e * H_HEADS + h] = a;
        atomicMax(&amaxkey[r * H_HEADS + h], f2key(a));
    }
}

// ---------------------------------------------------------------------------
// Kernel 4: edge pass 2 — a = exp(alpha - amax[row]); segment sum.
// ---------------------------------------------------------------------------
__global__ void edge_exp_kernel(const int* __restrict__ eidx, int E, int Etot,
                                float* __restrict__ alpha,
                                const int* __restrict__ amaxkey,
                                float* __restrict__ asum) {
    int e = blockIdx.x * blockDim.x + threadIdx.x;
    if (e >= Etot) return;
    int r = (e < E) ? eidx[e] : (e - E);
    #pragma unroll
    for (int h = 0; h < H_HEADS; ++h) {
        float m = key2f(amaxkey[r * H_HEADS + h]);
        if (!isfinite(m)) m = 0.0f;
        float a = __expf(alpha[(size_t)e * H_HEADS + h] - m);
        alpha[(size_t)e * H_HEADS + h] = a;
        atomicAdd(&asum[r * H_HEADS + h], a);
    }
}

// ---------------------------------------------------------------------------
// Kernel 5: edge pass 3 — alpha_sm = a/(asum+eps); Gram partials (LDS-reduced);
// scatter out[row] += content[col]*alpha_sm.
// ---------------------------------------------------------------------------
__global__ __launch_bounds__(256) void edge_scatter_kernel(
        const int* __restrict__ eidx, int E, int Etot,
        const float* __restrict__ content,
        const float* __restrict__ alpha,
        const float* __restrict__ asum,
        float* __restrict__ out,
        float* __restrict__ gram) {
    __shared__ float sG[10];
    if (threadIdx.x < 10) sG[threadIdx.x] = 0.0f;
    __syncthreads();

    int e = blockIdx.x * blockDim.x + threadIdx.x;
    if (e < Etot) {
        int r = (e < E) ? eidx[e]     : (e - E);
        int c = (e < E) ? eidx[E + e] : (e - E);
        float w[H_HEADS];
        #pragma unroll
        for (int h = 0; h < H_HEADS; ++h)
            w[h] = alpha[(size_t)e * H_HEADS + h]
                 / (asum[r * H_HEADS + h] + 1e-16f);

        // upper-triangular Gram partials:
        // (0,0)(0,1)(0,2)(0,3)(1,1)(1,2)(1,3)(2,2)(2,3)(3,3)
        int idx = 0;
        #pragma unroll
        for (int h1 = 0; h1 < H_HEADS; ++h1)
            #pragma unroll
            for (int h2 = h1; h2 < H_HEADS; ++h2)
                atomicAdd(&sG[idx++], w[h1] * w[h2]);

        const float* cj = content + (size_t)c * HC;
        float* dst = out + (size_t)r * HC;
        #pragma unroll
        for (int h = 0; h < H_HEADS; ++h) {
            float wh = w[h];
            #pragma unroll
            for (int k = 0; k < C_CH; ++k)
                atomicAdd(&dst[h * C_CH + k], cj[h * C_CH + k] * wh);
        }
    }
    __syncthreads();
    if (threadIdx.x < 10) atomicAdd(&gram[threadIdx.x], sG[threadIdx.x]);
}

// ---------------------------------------------------------------------------
// Kernel 6: diversity loss from the 10 Gram sums.
// ---------------------------------------------------------------------------
__global__ void finalize_kernel(const float* __restrict__ gram,
                                float* __restrict__ div_out) {
    if (blockIdx.x == 0 && threadIdx.x == 0) {
        const int diag[4] = {0, 4, 7, 9};
        float norm[4];
        #pragma unroll
        for (int h = 0; h < 4; ++h)
            norm[h] = fmaxf(sqrtf(gram[diag[h]]), 1e-8f);
        // off-diagonal packed indices: (0,1)=1 (0,2)=2 (0,3)=3 (1,2)=5 (1,3)=6 (2,3)=8
        const int oi[6]  = {1, 2, 3, 5, 6, 8};
        const int oh1[6] = {0, 0, 0, 1, 1, 2};
        const int oh2[6] = {1, 2, 3, 2, 3, 3};
        float total = 0.0f;
        #pragma unroll
        for (int p = 0; p < 6; ++p)
            total += 2.0f * gram[oi[p]] / (norm[oh1[p]] * norm[oh2[p]]);
        div_out[0] = (total / 16.0f) * DIV_W;
    }
}

// ---------------------------------------------------------------------------
extern "C" void kernel_launch(void* const* d_in, const int* in_sizes, int n_in,
                              void* d_out, int out_size, void* d_ws, size_t ws_size,
                              hipStream_t stream) {
    const float* x       = (const float*)d_in[0];
    const int*   eidx    = (const int*)  d_in[1];
    const float* lin_w   = (const float*)d_in[2];
    const float* att     = (const float*)d_in[3];
    const float* pos_att = (const float*)d_in[4];
    const float* bias    = (const float*)d_in[5];

    const int n_nodes = in_sizes[0] / ROWLEN;     // 100000
    const int E       = in_sizes[1] / 2;          // 1600000
    const int Etot    = E + n_nodes;              // edges + self loops

    float* out = (float*)d_out;                   // N*64 floats, then diversity

    // workspace layout
    char* w = (char*)d_ws;
    float* content = (float*)w;  w += (size_t)n_nodes * HC * sizeof(float);
    float* svec    = (float*)w;  w += (size_t)n_nodes * H_HEADS * sizeof(float);
    float* tvec    = (float*)w;  w += (size_t)n_nodes * H_HEADS * sizeof(float);
    float* alpha   = (float*)w;  w += (size_t)Etot * H_HEADS * sizeof(float);
    int*   amaxkey = (int*)w;    w += (size_t)n_nodes * H_HEADS * sizeof(int);
    float* asum    = (float*)w;  w += (size_t)n_nodes * H_HEADS * sizeof(float);
    float* gram    = (float*)w;  w += 16 * sizeof(float);
    (void)ws_size; (void)n_in; (void)out_size;

    const int T = 256;
    // 0: init accumulators
    {
        int total = n_nodes * HC;
        init_kernel<<<(total + T - 1) / T, T, 0, stream>>>(bias, out, amaxkey, asum, gram, n_nodes);
    }
    // 1: WMMA GEMM -> content
    {
        int nstrips = (n_nodes + 15) / 16;
        int blocks  = (nstrips + 3) / 4;          // 4 waves per block
        gemm_content_kernel<<<blocks, 128, 0, stream>>>(x, lin_w, content, n_nodes);
    }
    // 2: per-node attention scalars
    {
        int total = n_nodes * H_HEADS;
        node_st_kernel<<<(total + T - 1) / T, T, 0, stream>>>(x, content, att, pos_att, svec, tvec, n_nodes);
    }
    // 3-5: edge passes
    {
        int blocks = (Etot + T - 1) / T;
        edge_alpha_kernel  <<<blocks, T, 0, stream>>>(eidx, E, Etot, svec, tvec, alpha, amaxkey);
        edge_exp_kernel    <<<blocks, T, 0, stream>>>(eidx, E, Etot, alpha, amaxkey, asum);
        edge_scatter_kernel<<<blocks, T, 0, stream>>>(eidx, E, Etot, content, alpha, asum, out, gram);
    }
    // 6: diversity scalar
    finalize_kernel<<<1, 32, 0, stream>>>(gram, out + (size_t)n_nodes * HC);
}